// PointPillarsVoxelization_4337916970092
// MI455X (gfx1250) — compile-verified
//
#include <hip/hip_runtime.h>
#include <hip/hip_bf16.h>
#include <stdint.h>

// ---------------- problem constants (from the reference) ----------------
#define N_FEATS   5
#define GX        400
#define GY        400
#define GZ        1
#define NCELL     (GX * GY * GZ)          // 160000
#define MAX_VOX   60000
#define MAX_PTS   32
#define XMIN      (-50.0f)
#define YMIN      (-50.0f)
#define ZMIN      (-5.0f)
#define INVX      (4.0f)                  // 1/0.25 (exact)
#define INVY      (4.0f)
#define INVZ      (0.125f)                // 1/8.0 (exact)

#define OUT_VOX_ELEMS   (MAX_VOX * MAX_PTS * N_FEATS)   // 9,600,000
#define OUT_COORD_ELEMS (MAX_VOX * 3)                   //   180,000
#define OUT_NP_ELEMS    (MAX_VOX)                       //    60,000
#define OUT_TOTAL       (OUT_VOX_ELEMS + OUT_COORD_ELEMS + OUT_NP_ELEMS)

#define THREADS        256
#define WAVES_PER_BLK  (THREADS / 32)
#define SLICE_F        (32 * N_FEATS)     // floats per wave slice = 160
#define CELLS_PER_SCAN_BLK 1024
#define NSCAN_BLOCKS   ((NCELL + CELLS_PER_SCAN_BLK - 1) / CELLS_PER_SCAN_BLK) // 157

// ---------------- CDNA5 TDM (tensor_load_to_lds) helpers ----------------
typedef unsigned int uint_t;
typedef uint_t v4u __attribute__((ext_vector_type(4)));
typedef int    v8i __attribute__((ext_vector_type(8)));
typedef int    v4i __attribute__((ext_vector_type(4)));

#if defined(__has_builtin)
#if __has_builtin(__builtin_amdgcn_tensor_load_to_lds)
#define HAVE_TDM 1
#endif
#if __has_builtin(__builtin_amdgcn_s_wait_tensorcnt)
#define HAVE_TENSORCNT 1
#endif
#endif

// Use the CDNA5 async LDS->memory store pipeline (ASYNCcnt) for the
// feature scatter in k_scatter. Data goes straight from LDS (where the TDM
// staged it) to per-lane global addresses, bypassing VGPRs.
#define ASYNC_SCATTER 1

#if defined(HAVE_TDM)
// 1-D DMA of `nvalid` f32 elements starting at `gsrc` into LDS byte offset
// `lds_off`. Tile is fixed at SLICE_F elements; the TDM zero-fills reads
// beyond tensor_dim0 (tail of the points array).
__device__ __forceinline__ void tdm_load_slice(const float* gsrc,
                                               uint_t lds_off,
                                               uint_t nvalid) {
  unsigned long long ga = (unsigned long long)(uintptr_t)gsrc;
  v4u g0;
  g0[0] = 1u;                                   // count=1, user mode
  g0[1] = lds_off;                              // lds_addr (bytes)
  g0[2] = (uint_t)(ga & 0xFFFFFFFFu);           // global_addr[31:0]
  g0[3] = (uint_t)((ga >> 32) & 0x01FFFFFFu)    // global_addr[56:32]
        | (2u << 30);                           // type = 2 ("image")
  const uint_t td0 = nvalid;                    // tensor_dim0 (f32 units)
  const uint_t td1 = 1u;                        // tensor_dim1
  const uint_t t0  = SLICE_F;                   // tile_dim0 = 160
  const uint_t s0  = SLICE_F;                   // tensor_dim0_stride
  const uint_t s1  = SLICE_F;                   // tensor_dim1_stride
  v8i g1;
  g1[0] = (int)(2u << 16);                             // data_size = 4B
  g1[1] = (int)((td0 & 0xFFFFu) << 16);                // tensor_dim0 lo16
  g1[2] = (int)((td0 >> 16) | ((td1 & 0xFFFFu) << 16));// dim0 hi16 | dim1 lo16
  g1[3] = (int)((td1 >> 16) | (t0 << 16));             // dim1 hi16 | tile_dim0
  g1[4] = 0;                                           // tile_dim1=0, tile_dim2=0
  g1[5] = (int)s0;                                     // dim0_stride[31:0]
  g1[6] = (int)(((s1 & 0xFFFFu) << 16));               // dim0_str hi16=0 | dim1_str lo16
  g1[7] = (int)(s1 >> 16);                             // dim1_stride[47:16]
  v4i z4 = {0, 0, 0, 0};
#if __clang_major__ >= 23
  v8i z8 = {0, 0, 0, 0, 0, 0, 0, 0};
  __builtin_amdgcn_tensor_load_to_lds(g0, g1, z4, z4, z8, 0);
#else
  __builtin_amdgcn_tensor_load_to_lds(g0, g1, z4, z4, 0);
#endif
}
#endif

__device__ __forceinline__ void wait_tensor0() {
#if defined(HAVE_TENSORCNT)
  __builtin_amdgcn_s_wait_tensorcnt(0);
#else
  asm volatile("s_wait_tensorcnt 0" ::: "memory");
#endif
  asm volatile("" ::: "memory");   // keep LDS reads after the wait
}

// Stage this wave's 32-point slice into its private LDS region.
// Returns pointer to the wave's 160-float LDS slice.
__device__ __forceinline__ const float* stage_wave_slice(const float* __restrict__ pts,
                                                         int block_base_pt,
                                                         int total_floats,
                                                         float* sbuf) {
  const int wave = (int)(threadIdx.x >> 5);
  const int lane = (int)(threadIdx.x & 31);
  int slice_f = (block_base_pt + wave * 32) * N_FEATS;
  float* lds_wave = sbuf + wave * SLICE_F;
#if defined(HAVE_TDM)
  int remaining = total_floats - slice_f;
  uint_t nvalid = remaining > 0 ? (uint_t)remaining : 0u;
  if (remaining <= 0) slice_f = 0;                    // keep addr in-buffer
  uint_t lds_off = (uint_t)(uintptr_t)lds_wave;       // flat->LDS: addr[31:0]
  tdm_load_slice(pts + slice_f, lds_off, nvalid);
  wait_tensor0();
  (void)lane;
#else
  // Fallback: each lane copies its own 5 floats (wave-private slots, no sync)
  int f0 = slice_f + lane * N_FEATS;
#pragma unroll
  for (int k = 0; k < N_FEATS; ++k)
    lds_wave[lane * N_FEATS + k] = (f0 + k < total_floats) ? pts[f0 + k] : 0.0f;
#endif
  return lds_wave;
}

__device__ __forceinline__ bool point_to_cell(float x, float y, float z, int* cell) {
  int cx = (int)floorf((x - XMIN) * INVX);
  int cy = (int)floorf((y - YMIN) * INVY);
  int cz = (int)floorf((z - ZMIN) * INVZ);
  bool ok = (cx >= 0) & (cx < GX) & (cy >= 0) & (cy < GY) & (cz >= 0) & (cz < GZ);
  *cell = cz * (GY * GX) + cy * GX + cx;
  return ok;
}

// ---------------- kernels ----------------

// K0: zero d_out + count[] + wcnt[] with 16B stores.
__global__ __launch_bounds__(THREADS) void k_zero(int4* __restrict__ out4,
                                                  int4* __restrict__ count4,
                                                  int4* __restrict__ wcnt4) {
  const int OUT4 = OUT_TOTAL / 4;     // 2,460,000
  const int CNT4 = NCELL / 4;         //    40,000
  const int WC4  = MAX_VOX / 4;       //    15,000
  int idx = (int)(blockIdx.x * blockDim.x + threadIdx.x);
  int4 zv; zv.x = zv.y = zv.z = zv.w = 0;
  if (idx < OUT4)                out4[idx] = zv;
  else if (idx < OUT4 + CNT4)    count4[idx - OUT4] = zv;
  else if (idx < OUT4 + CNT4 + WC4) wcnt4[idx - OUT4 - CNT4] = zv;
}

// K1: per-cell point histogram.
__global__ __launch_bounds__(THREADS) void k_count(const float* __restrict__ pts,
                                                   int npts, int total_floats,
                                                   int* __restrict__ count) {
  __shared__ float sbuf[WAVES_PER_BLK * SLICE_F];
  int base = (int)(blockIdx.x * blockDim.x);
  const float* my = stage_wave_slice(pts, base, total_floats, sbuf);
  int lane = (int)(threadIdx.x & 31);
  int i = base + (int)threadIdx.x;
  if (i >= npts) return;
  float x = my[lane * N_FEATS + 0];
  float y = my[lane * N_FEATS + 1];
  float z = my[lane * N_FEATS + 2];
  int cell;
  if (point_to_cell(x, y, z, &cell)) atomicAdd(&count[cell], 1);
}

// K2a: per-scan-block sum of occupancy flags.
__global__ __launch_bounds__(THREADS) void k_scan_partial(const int* __restrict__ count,
                                                          int* __restrict__ blockSums) {
  __shared__ int red[THREADS];
  int blk = (int)blockIdx.x, t = (int)threadIdx.x;
  int cbase = blk * CELLS_PER_SCAN_BLK + t * 4;
  int s = 0;
#pragma unroll
  for (int k = 0; k < 4; ++k) {
    int c = cbase + k;
    if (c < NCELL) s += (count[c] > 0);
  }
  red[t] = s;
  __syncthreads();
  for (int off = THREADS / 2; off > 0; off >>= 1) {
    if (t < off) red[t] += red[t + off];
    __syncthreads();
  }
  if (t == 0) blockSums[blk] = red[0];
}

// K2b: serial exclusive scan of 157 block sums (trivial).
__global__ void k_scan_blocks(int* __restrict__ blockSums, int nblk) {
  if (blockIdx.x == 0 && threadIdx.x == 0) {
    int run = 0;
    for (int i = 0; i < nblk; ++i) {
      int v = blockSums[i];
      blockSums[i] = run;
      run += v;
    }
  }
}

// K2c: assign voxel ids in ascending flat-cell order; emit coords + num_points.
__global__ __launch_bounds__(THREADS) void k_assign(const int* __restrict__ count,
                                                    const int* __restrict__ blockSums,
                                                    int* __restrict__ voxmap,
                                                    int* __restrict__ out_coords,
                                                    int* __restrict__ out_np) {
  __shared__ int sc[THREADS];
  int blk = (int)blockIdx.x, t = (int)threadIdx.x;
  int cbase = blk * CELLS_PER_SCAN_BLK + t * 4;
  int occ[4];
  int s = 0;
#pragma unroll
  for (int k = 0; k < 4; ++k) {
    int c = cbase + k;
    occ[k] = (c < NCELL && count[c] > 0) ? 1 : 0;
    s += occ[k];
  }
  sc[t] = s;
  __syncthreads();
  for (int off = 1; off < THREADS; off <<= 1) {
    int v = (t >= off) ? sc[t - off] : 0;
    __syncthreads();
    sc[t] += v;
    __syncthreads();
  }
  int prefix = blockSums[blk] + (sc[t] - s);   // exclusive prefix for this thread
#pragma unroll
  for (int k = 0; k < 4; ++k) {
    int c = cbase + k;
    if (c >= NCELL) continue;
    if (occ[k]) {
      if (prefix < MAX_VOX) {
        voxmap[c] = prefix;
        int cz = c / (GX * GY);
        int rem = c % (GX * GY);
        int cy = rem / GX;
        int cx = rem % GX;
        out_coords[prefix * 3 + 0] = cz;   // zyx order
        out_coords[prefix * 3 + 1] = cy;
        out_coords[prefix * 3 + 2] = cx;
        int cnt = count[c];
        out_np[prefix] = cnt < MAX_PTS ? cnt : MAX_PTS;
      } else {
        voxmap[c] = -1;
      }
      prefix++;
    } else {
      voxmap[c] = -1;
    }
  }
}

// K3: scatter point features into out_voxels.
// With ASYNC_SCATTER, only x/y/z are read into VGPRs for the cell compute;
// all 5 features are shipped LDS -> global via the async store pipeline.
__global__ __launch_bounds__(THREADS) void k_scatter(const float* __restrict__ pts,
                                                     int npts, int total_floats,
                                                     const int* __restrict__ voxmap,
                                                     int* __restrict__ wcnt,
                                                     float* __restrict__ out_vox) {
  __shared__ float sbuf[WAVES_PER_BLK * SLICE_F];
  int base = (int)(blockIdx.x * blockDim.x);
  const float* my = stage_wave_slice(pts, base, total_floats, sbuf);
  int lane = (int)(threadIdx.x & 31);
  int i = base + (int)threadIdx.x;
  if (i >= npts) return;
  const float* p = my + lane * N_FEATS;
  float x = p[0], y = p[1], z = p[2];
  int cell;
  if (!point_to_cell(x, y, z, &cell)) return;
  int v = voxmap[cell];
  if (v < 0) return;
  int r = atomicAdd(&wcnt[v], 1);
  if (r >= MAX_PTS) return;
  float* dst = out_vox + ((size_t)v * MAX_PTS + (size_t)r) * N_FEATS;
#if ASYNC_SCATTER
  // Per-lane async stores: LDS source (4B aligned, lane*20 + k*4) -> global.
  // Tracked by ASYNCcnt; S_ENDPGM's implicit wait-idle covers completion and
  // the LDS slice is never overwritten during the kernel.
  unsigned long long ga = (unsigned long long)(uintptr_t)dst;
  uint_t lds0 = (uint_t)(uintptr_t)p;   // flat->LDS: addr[31:0]
#pragma unroll
  for (int k = 0; k < N_FEATS; ++k) {
    unsigned long long gak = ga + (unsigned long long)(k * 4);
    uint_t ldsk = lds0 + (uint_t)(k * 4);
    asm volatile("global_store_async_from_lds_b32 %0, %1, off"
                 :: "v"(gak), "v"(ldsk)
                 : "memory");
  }
#else
  float p3 = p[3], p4 = p[4];
  dst[0] = x; dst[1] = y; dst[2] = z; dst[3] = p3; dst[4] = p4;
#endif
}

// ---------------- host launcher ----------------
extern "C" void kernel_launch(void* const* d_in, const int* in_sizes, int n_in,
                              void* d_out, int out_size, void* d_ws, size_t ws_size,
                              hipStream_t stream) {
  const float* pts = (const float*)d_in[0];
  const int total_floats = in_sizes[0];
  const int npts = total_floats / N_FEATS;

  // workspace layout (ints): count[NCELL] | voxmap[NCELL] | wcnt[MAX_VOX] | blockSums
  int* ws        = (int*)d_ws;
  int* count     = ws;
  int* voxmap    = ws + NCELL;
  int* wcnt      = ws + 2 * NCELL;
  int* blockSums = ws + 2 * NCELL + MAX_VOX;

  float* out_vox    = (float*)d_out;
  int*   out_coords = (int*)d_out + OUT_VOX_ELEMS;
  int*   out_np     = out_coords + OUT_COORD_ELEMS;

  const int zero4 = OUT_TOTAL / 4 + NCELL / 4 + MAX_VOX / 4;
  const int zb = (zero4 + THREADS - 1) / THREADS;
  k_zero<<<zb, THREADS, 0, stream>>>((int4*)d_out, (int4*)count, (int4*)wcnt);

  const int nb = (npts + THREADS - 1) / THREADS;
  k_count<<<nb, THREADS, 0, stream>>>(pts, npts, total_floats, count);
  k_scan_partial<<<NSCAN_BLOCKS, THREADS, 0, stream>>>(count, blockSums);
  k_scan_blocks<<<1, 32, 0, stream>>>(blockSums, NSCAN_BLOCKS);
  k_assign<<<NSCAN_BLOCKS, THREADS, 0, stream>>>(count, blockSums, voxmap,
                                                 out_coords, out_np);
  k_scatter<<<nb, THREADS, 0, stream>>>(pts, npts, total_floats, voxmap, wcnt,
                                        out_vox);
}